// Net_40312563041045
// MI455X (gfx1250) — compile-verified
//
#include <hip/hip_runtime.h>

typedef __attribute__((ext_vector_type(2))) float v2f;
typedef __attribute__((ext_vector_type(4))) float v4f;
typedef __attribute__((ext_vector_type(8))) float v8f;

#define BETA 0.2f

constexpr int KC = 32;       // K chunk
constexpr int LDR = KC + 4;  // padded LDS row stride (floats): 36 -> 16B aligned,
                             // gcd(36,64)=4 -> 16 distinct banks for 16 rows

// ---- CDNA5 async global->LDS staging (ASYNCcnt-tracked), with sync fallback ----
#if defined(__AMDGCN__) && __has_builtin(__builtin_amdgcn_global_load_async_to_lds_b128)
#define ASYNC_LDS 1
#else
#define ASYNC_LDS 0
#endif

// The ROCm clang builtin signature (from the round-2 diagnostic) takes pointers
// to GCC-vector 'int __attribute__((vector_size(16)))' in AS1 (global) / AS3 (LDS).
typedef int b128_t __attribute__((vector_size(16)));

__device__ __forceinline__ void copy16_to_lds(const float* g, float* l) {
#if ASYNC_LDS
  __builtin_amdgcn_global_load_async_to_lds_b128(
      (__attribute__((address_space(1))) b128_t*)(void*)g,
      (__attribute__((address_space(3))) b128_t*)(void*)l, 0, 0);
#else
  *(v4f*)l = *(const v4f*)g;
#endif
}

__device__ __forceinline__ void wait_async_lds() {
#if ASYNC_LDS
#if __has_builtin(__builtin_amdgcn_s_wait_asynccnt)
  __builtin_amdgcn_s_wait_asynccnt(0);
#else
  asm volatile("s_wait_asynccnt 0" ::: "memory");
#endif
#endif
}

// Stage one K-chunk: X tile 32xKC, W tile 64xKC into (padded) LDS buffers.
template <int KDIM>
__device__ __forceinline__ void stage_chunk(const float* __restrict__ X,
                                            const float* __restrict__ W,
                                            int m0, int n0, int k0, int t,
                                            float* __restrict__ Xb,
                                            float* __restrict__ Wb) {
  const int row = t >> 3;       // 0..31
  const int c4 = (t & 7) << 2;  // 0,4,..,28
  copy16_to_lds(&X[(m0 + row) * KDIM + k0 + c4], &Xb[row * LDR + c4]);
#pragma unroll
  for (int r = 0; r < 2; ++r) {
    const int s = t + (r << 8);  // 0..511
    const int wr = s >> 3;       // 0..63
    const int wc = (s & 7) << 2;
    copy16_to_lds(&W[(n0 + wr) * KDIM + k0 + wc], &Wb[wr * LDR + wc]);
  }
}

// Fused "max-tempered" layer:
//   Y[m,n] = (1-BETA)*sum_i W[n,i]*X[m,i] + BETA*max_i W[n,i]*X[m,i] + bias[n]
//
// Block = 256 threads = 8 wave32s; block tile M=32 x N=64; wave tile 16x16.
// Double-buffered LDS: chunk c+1 streams in via GLOBAL_LOAD_ASYNC_TO_LDS_B128
// while chunk c feeds the fp32 WMMA (linear part) and the VALU pk_mul/max3
// stream (tropical part). Lane ownership of both accumulators matches the
// WMMA C/D layout (N = lane%16, M = j + 8*(lane/16)), so no final shuffles.
template <int KDIM>
__global__ __launch_bounds__(256) void tropical_layer(
    const float* __restrict__ X,     // [B, KDIM]
    const float* __restrict__ W,     // [NOUT, KDIM] row-major
    const float* __restrict__ bias,  // [NOUT]
    float* __restrict__ Y,           // [B, NOUT]
    int NOUT) {
  __shared__ float Xs[2][32 * LDR];
  __shared__ float Ws[2][64 * LDR];

  const int t = threadIdx.x;
  const int lane = t & 31;
  const int wave = t >> 5;
  const int lo = lane & 15;
  const int hi = lane >> 4;
  const int mt = (wave >> 2) << 4;  // 0 or 16
  const int nt = (wave & 3) << 4;   // 0,16,32,48
  const int m0 = blockIdx.x * 32;
  const int n0 = blockIdx.y * 64;

  v8f acc = {};  // WMMA C/D accumulator (16x16 f32, 8 VGPRs)
  float mx[8];
#pragma unroll
  for (int j = 0; j < 8; ++j) mx[j] = -__builtin_inff();

  constexpr int NC = KDIM / KC;

  // prologue: stage chunk 0 into buffer 0
  stage_chunk<KDIM>(X, W, m0, n0, 0, t, Xs[0], Ws[0]);
  wait_async_lds();
  __syncthreads();

  for (int c = 0; c < NC; ++c) {
    const int cur = c & 1;
    // prefetch chunk c+1 into the other buffer (fully consumed at the barrier
    // that ended iteration c-1, so it is free to overwrite)
    if (c + 1 < NC)
      stage_chunk<KDIM>(X, W, m0, n0, (c + 1) * KC, t, Xs[cur ^ 1],
                        Ws[cur ^ 1]);

    const float* __restrict__ xb = Xs[cur];
    const float* __restrict__ wb = Ws[cur];

    // ---- linear part: fp32 WMMA 16x16x4, K advanced 4 per instruction ----
    // A frag (32-bit A 16x4): lane<16 holds K={kk,kk+1}, lane>=16 K={kk+2,kk+3},
    // row M = lane%16 -> one 8-byte LDS read per step; B mirrored from W rows.
#pragma unroll
    for (int kk = 0; kk < KC; kk += 4) {
      v2f a = *(const v2f*)&xb[(mt + lo) * LDR + kk + 2 * hi];
      v2f b = *(const v2f*)&wb[(nt + lo) * LDR + kk + 2 * hi];
      acc = __builtin_amdgcn_wmma_f32_16x16x4_f32(false, a, false, b, (short)0,
                                                  acc, false, false);
    }

    // ---- tropical part: VALU mul+max (no matrix unit can do this) ----
    // X reads are half-wave broadcasts (conflict-free); W reads hit 16 banks.
#pragma unroll
    for (int ii = 0; ii < KC; ii += 4) {
      v4f wv = *(const v4f*)&wb[(nt + lo) * LDR + ii];
#pragma unroll
      for (int j = 0; j < 8; ++j) {
        v4f xv = *(const v4f*)&xb[(mt + 8 * hi + j) * LDR + ii];
        mx[j] = fmaxf(mx[j], wv.x * xv.x);
        mx[j] = fmaxf(mx[j], wv.y * xv.y);
        mx[j] = fmaxf(mx[j], wv.z * xv.z);
        mx[j] = fmaxf(mx[j], wv.w * xv.w);
      }
    }

    // chunk c+1 must have fully landed before anyone reads it next iteration
    wait_async_lds();
    __syncthreads();
  }

  // ---- blend + bias + store (ownership matches C/D layout, no shuffles) ----
  const float bs = bias[n0 + nt + lo];
#pragma unroll
  for (int j = 0; j < 8; ++j) {
    const int m = m0 + mt + 8 * hi + j;
    Y[m * NOUT + n0 + nt + lo] = (1.0f - BETA) * acc[j] + BETA * mx[j] + bs;
  }
}

// Final layer: OUT=1. One wave32 per batch row; shuffle reduction.
__global__ __launch_bounds__(256) void tropical_out(
    const float* __restrict__ H,   // [B, 512]
    const float* __restrict__ W4,  // [512]
    const float* __restrict__ b4,  // [1]
    float* __restrict__ out) {     // [B]
  const int b = blockIdx.x * 8 + (threadIdx.x >> 5);
  const int lane = threadIdx.x & 31;
  const float* h = H + b * 512;
  float acc = 0.0f;
  float mx = -__builtin_inff();
#pragma unroll 4
  for (int i = lane; i < 512; i += 32) {
    const float p = W4[i] * h[i];
    acc += p;
    mx = fmaxf(mx, p);
  }
#pragma unroll
  for (int off = 16; off > 0; off >>= 1) {
    acc += __shfl_xor(acc, off, 32);
    mx = fmaxf(mx, __shfl_xor(mx, off, 32));
  }
  if (lane == 0) out[b] = (1.0f - BETA) * acc + BETA * mx + b4[0];
}

extern "C" void kernel_launch(void* const* d_in, const int* in_sizes, int n_in,
                              void* d_out, int out_size, void* d_ws,
                              size_t ws_size, hipStream_t stream) {
  const float* x = (const float*)d_in[0];
  const float* W1 = (const float*)d_in[1];
  const float* b1 = (const float*)d_in[2];
  const float* W2 = (const float*)d_in[3];
  const float* b2 = (const float*)d_in[4];
  const float* W3 = (const float*)d_in[5];
  const float* b3 = (const float*)d_in[6];
  const float* W4 = (const float*)d_in[7];
  const float* b4 = (const float*)d_in[8];

  // scratch: three 1024x512 fp32 activation buffers (6 MB)
  float* h1 = (float*)d_ws;
  float* h2 = h1 + 1024 * 512;
  float* h3 = h2 + 1024 * 512;

  const dim3 grid(32, 8);  // (1024/32) x (512/64)
  const dim3 block(256);
  tropical_layer<256><<<grid, block, 0, stream>>>(x, W1, b1, h1, 512);
  tropical_layer<512><<<grid, block, 0, stream>>>(h1, W2, b2, h2, 512);
  tropical_layer<512><<<grid, block, 0, stream>>>(h2, W3, b3, h3, 512);
  tropical_out<<<128, 256, 0, stream>>>(h3, W4, b4, (float*)d_out);
}